// GraphChebNetWithCoarsening_37400575213799
// MI455X (gfx1250) — compile-verified
//
#include <hip/hip_runtime.h>
#include <math.h>

#define N_NODES 100000
#define N_EDGES 1600000
#define F_INC   64
#define HID     128
#define NCLS    10
#define NGRAPH  64

typedef __attribute__((ext_vector_type(2))) float v2f;
typedef __attribute__((ext_vector_type(8))) float v8f;

__device__ __forceinline__ void atomAddF(float* p, float v) {
  unsafeAtomicAdd(p, v);   // hardware global_atomic_add_f32 on gfx1250
}

// ---------- degree: deg[src] += (src!=dst) ----------
__global__ void k_deg(const int* __restrict__ src, const int* __restrict__ dst,
                      float* __restrict__ deg, int nE, int shift) {
  int e = blockIdx.x * blockDim.x + threadIdx.x;
  if (e >= nE) return;
  int s = src[e] >> shift, d = dst[e] >> shift;
  if (s != d) atomAddF(&deg[s], 1.0f);
}

// ---------- deg -> rsqrt(deg) in place ----------
__global__ void k_dinv(float* __restrict__ deg, int n) {
  int i = blockIdx.x * blockDim.x + threadIdx.x;
  if (i >= n) return;
  float d = deg[i];
  deg[i] = (d > 0.0f) ? rsqrtf(d) : 0.0f;
}

// ---------- w[e] = -(dinv[s]*dinv[d]) for non-self-loops ----------
__global__ void k_edgew(const int* __restrict__ src, const int* __restrict__ dst,
                        const float* __restrict__ dinv, float* __restrict__ w,
                        int nE, int shift) {
  int e = blockIdx.x * blockDim.x + threadIdx.x;
  if (e >= nE) return;
  int s = src[e] >> shift, d = dst[e] >> shift;
  w[e] = (s != d) ? -(dinv[s] * dinv[d]) : 0.0f;
}

// ---------- SpMM: y[dst][:] += w[e] * z[src][:]  (one WAVE per edge) ----------
// Edge scalars are wave-uniform (scalarizable); feature loads/atomics are
// lane-coalesced. Zero-weight (self-loop) edges skip without divergence.
template<int LOGF>
__global__ void k_prop(const int* __restrict__ src, const int* __restrict__ dst,
                       const float* __restrict__ w, const float* __restrict__ z,
                       float* __restrict__ y, int nE, int shift) {
  const int lane = threadIdx.x & 31;
  const unsigned e = (unsigned)blockIdx.x * (blockDim.x >> 5) + (threadIdx.x >> 5);
  if (e >= (unsigned)nE) return;
  const float we = w[e];
  if (we == 0.0f) return;
  const size_t srow = (size_t)(src[e] >> shift) << LOGF;
  const size_t drow = (size_t)(dst[e] >> shift) << LOGF;
  constexpr int F = 1 << LOGF;
#pragma unroll
  for (int f = lane; f < F; f += 32)
    atomAddF(&y[drow + f], we * z[srow + f]);
}

// ---------- p = 2*p - x0 (Chebyshev recurrence finalize) ----------
__global__ void k_cheb2(float* __restrict__ p, const float* __restrict__ x0, size_t n) {
  size_t i = (size_t)blockIdx.x * blockDim.x + threadIdx.x;
  if (i >= n) return;
  p[i] = 2.0f * p[i] - x0[i];
}

// ---------- fused 3-term Cheb GEMM + bias + ReLU via V_WMMA_F32_16X16X4_F32 ----------
// out[n_rows,HID] = relu(A0@W[0] + A1@W[1] + A2@W[2] + bias)
// One wave computes a 16x64 output strip (4 WMMA tiles): each A fragment is
// reused by 4 WMMAs, and 4 independent accumulators keep the XDL pipe busy.
template<int K>
__global__ void k_cheb_gemm_relu(const float* __restrict__ A0,
                                 const float* __restrict__ A1,
                                 const float* __restrict__ A2,
                                 const float* __restrict__ W,    // [3][K][HID]
                                 const float* __restrict__ bias, // [HID]
                                 float* __restrict__ out,        // [n_rows][HID]
                                 int n_rows) {
  const int lane = threadIdx.x & 31;
  const int wave = threadIdx.x >> 5;
  const int wid  = blockIdx.x * (blockDim.x >> 5) + wave;
  const int ngrp = HID / 64;                    // 2 strips of 64 cols
  const int mt = wid / ngrp;
  const int ng = wid - mt * ngrp;
  const int m0 = mt * 16;
  if (m0 >= n_rows) return;

  const int half = lane >> 4;                   // 0: K={0,1}, 1: K={2,3}
  const int l15  = lane & 15;
  const int row  = m0 + l15;
  const int col0 = ng * 64 + l15;

  v8f acc0 = {}, acc1 = {}, acc2 = {}, acc3 = {};
  const float* Aps[3] = {A0, A1, A2};
#pragma unroll
  for (int p = 0; p < 3; ++p) {
    const float* Ap = Aps[p] + (size_t)row * K;
    const float* Wp = W + (size_t)p * K * HID + col0;
#pragma unroll 4
    for (int k0 = 0; k0 < K; k0 += 4) {
      const int kk = k0 + (half << 1);
      v2f a;
      a.x = Ap[kk];
      a.y = Ap[kk + 1];
      const float* wr0 = Wp + (size_t)kk * HID;
      const float* wr1 = Wp + (size_t)(kk + 1) * HID;
      v2f b0, b1, b2, b3;
      b0.x = wr0[0];  b0.y = wr1[0];
      b1.x = wr0[16]; b1.y = wr1[16];
      b2.x = wr0[32]; b2.y = wr1[32];
      b3.x = wr0[48]; b3.y = wr1[48];
      acc0 = __builtin_amdgcn_wmma_f32_16x16x4_f32(false, a, false, b0, (short)0, acc0, false, false);
      acc1 = __builtin_amdgcn_wmma_f32_16x16x4_f32(false, a, false, b1, (short)0, acc1, false, false);
      acc2 = __builtin_amdgcn_wmma_f32_16x16x4_f32(false, a, false, b2, (short)0, acc2, false, false);
      acc3 = __builtin_amdgcn_wmma_f32_16x16x4_f32(false, a, false, b3, (short)0, acc3, false, false);
    }
  }

  const float bv0 = bias[col0];
  const float bv1 = bias[col0 + 16];
  const float bv2 = bias[col0 + 32];
  const float bv3 = bias[col0 + 48];
#pragma unroll
  for (int v = 0; v < 8; ++v) {
    float* orow = out + (size_t)(m0 + v + half * 8) * HID + col0;
    orow[0]  = fmaxf(acc0[v] + bv0, 0.0f);
    orow[16] = fmaxf(acc1[v] + bv1, 0.0f);
    orow[32] = fmaxf(acc2[v] + bv2, 0.0f);
    orow[48] = fmaxf(acc3[v] + bv3, 0.0f);
  }
}

// ---------- h2[i] = 0.5*(h[2i] + h[2i+1]) row-pair mean ----------
__global__ void k_coarsen(const float* __restrict__ h, float* __restrict__ h2, int n2) {
  size_t t = (size_t)blockIdx.x * blockDim.x + threadIdx.x;
  if (t >= (size_t)n2 * HID) return;
  size_t i = t >> 7;                            // HID == 128
  int    f = (int)(t & 127);
  h2[t] = 0.5f * (h[((i * 2) << 7) + f] + h[((i * 2 + 1) << 7) + f]);
}

// ---------- segment-sum pool over graphs (one WAVE per node) ----------
__global__ void k_pool(const float* __restrict__ h2, const int* __restrict__ batch,
                       float* __restrict__ sums, float* __restrict__ cnts, int n2) {
  const int lane = threadIdx.x & 31;
  const unsigned i = (unsigned)blockIdx.x * (blockDim.x >> 5) + (threadIdx.x >> 5);
  if (i >= (unsigned)n2) return;
  const int g = batch[2 * i];                   // batch2[i] = batch[2i]
  const size_t base = (size_t)i << 7;
  float* srow = &sums[(size_t)g * HID];
#pragma unroll
  for (int f = lane; f < HID; f += 32)
    atomAddF(&srow[f], h2[base + f]);
  if (lane == 0) atomAddF(&cnts[g], 1.0f);
}

// ---------- pooled = sums/max(cnt,1); logits = pooled@fcW + fcb; log_softmax ----------
__global__ void k_head(const float* __restrict__ sums, const float* __restrict__ cnts,
                       const float* __restrict__ fcW, const float* __restrict__ fcb,
                       float* __restrict__ out) {
  __shared__ float lg[NGRAPH * NCLS];
  const int t = threadIdx.x;                    // 0..639
  const int g = t / NCLS;
  const int c = t - g * NCLS;
  const float inv = 1.0f / fmaxf(cnts[g], 1.0f);
  float acc = fcb[c];
  for (int h = 0; h < HID; ++h)
    acc += (sums[g * HID + h] * inv) * fcW[h * NCLS + c];
  lg[t] = acc;
  __syncthreads();
  float m = -3.402823466e38f;
  for (int j = 0; j < NCLS; ++j) m = fmaxf(m, lg[g * NCLS + j]);
  float s = 0.0f;
  for (int j = 0; j < NCLS; ++j) s += expf(lg[g * NCLS + j] - m);
  out[t] = acc - m - logf(s);
}

extern "C" void kernel_launch(void* const* d_in, const int* in_sizes, int n_in,
                              void* d_out, int out_size, void* d_ws, size_t ws_size,
                              hipStream_t stream) {
  const float* x    = (const float*)d_in[0];
  const int*   ei   = (const int*)d_in[1];
  const int*   batch= (const int*)d_in[2];
  const float* W1   = (const float*)d_in[3];
  const float* b1   = (const float*)d_in[4];
  const float* W2   = (const float*)d_in[5];
  const float* b2   = (const float*)d_in[6];
  const float* fcW  = (const float*)d_in[7];
  const float* fcb  = (const float*)d_in[8];
  float* out = (float*)d_out;

  const int* src = ei;
  const int* dst = ei + N_EDGES;

  // workspace layout (floats)
  float* wsf = (float*)d_ws;
  size_t off = 0;
  float* w    = wsf + off; off += N_EDGES;                       // 1.6M
  float* deg  = wsf + off; off += N_NODES;                       // 0.1M
  float* bufA = wsf + off; off += (size_t)N_NODES * F_INC;       // 6.4M (== (N/2)*HID)
  float* bufB = wsf + off; off += (size_t)N_NODES * F_INC;       // 6.4M
  float* h    = wsf + off; off += (size_t)N_NODES * HID;         // 12.8M
  float* h2   = wsf + off; off += (size_t)(N_NODES / 2) * HID;   // 6.4M
  float* sums = wsf + off; off += (size_t)NGRAPH * HID;
  float* cnts = wsf + off; off += NGRAPH;
  float* out2 = h;  // layer-2 output reuses h (h no longer needed after coarsen)

  const int TB = 256;                   // 8 waves per block
  auto cdiv = [](size_t a, size_t b) { return (unsigned)((a + b - 1) / b); };

  // ===== layer 1: scaled-Laplacian edge weights =====
  hipMemsetAsync(deg, 0, N_NODES * sizeof(float), stream);
  k_deg  <<<cdiv(N_EDGES, TB), TB, 0, stream>>>(src, dst, deg, N_EDGES, 0);
  k_dinv <<<cdiv(N_NODES, TB), TB, 0, stream>>>(deg, N_NODES);
  k_edgew<<<cdiv(N_EDGES, TB), TB, 0, stream>>>(src, dst, deg, w, N_EDGES, 0);

  // Tx1 = L~ x ; Tx2 = 2 L~ Tx1 - x   (wave-per-edge SpMM)
  hipMemsetAsync(bufA, 0, (size_t)N_NODES * F_INC * sizeof(float), stream);
  k_prop<6><<<cdiv(N_EDGES, 8), TB, 0, stream>>>(src, dst, w, x, bufA, N_EDGES, 0);
  hipMemsetAsync(bufB, 0, (size_t)N_NODES * F_INC * sizeof(float), stream);
  k_prop<6><<<cdiv(N_EDGES, 8), TB, 0, stream>>>(src, dst, w, bufA, bufB, N_EDGES, 0);
  k_cheb2<<<cdiv((size_t)N_NODES * F_INC, TB), TB, 0, stream>>>(bufB, x, (size_t)N_NODES * F_INC);

  // h = relu(x@W1[0] + Tx1@W1[1] + Tx2@W1[2] + b1)   [WMMA, 16x64 strips]
  {
    const unsigned waves = (N_NODES / 16) * (HID / 64);   // 12500
    k_cheb_gemm_relu<F_INC><<<cdiv(waves, 4), 128, 0, stream>>>(x, bufA, bufB, W1, b1, h, N_NODES);
  }

  // coarsen pairs
  k_coarsen<<<cdiv((size_t)(N_NODES / 2) * HID, TB), TB, 0, stream>>>(h, h2, N_NODES / 2);

  // ===== layer 2 on coarsened graph (indices >> 1) =====
  hipMemsetAsync(deg, 0, (N_NODES / 2) * sizeof(float), stream);
  k_deg  <<<cdiv(N_EDGES, TB), TB, 0, stream>>>(src, dst, deg, N_EDGES, 1);
  k_dinv <<<cdiv(N_NODES / 2, TB), TB, 0, stream>>>(deg, N_NODES / 2);
  k_edgew<<<cdiv(N_EDGES, TB), TB, 0, stream>>>(src, dst, deg, w, N_EDGES, 1);

  hipMemsetAsync(bufA, 0, (size_t)(N_NODES / 2) * HID * sizeof(float), stream);
  k_prop<7><<<cdiv(N_EDGES, 8), TB, 0, stream>>>(src, dst, w, h2, bufA, N_EDGES, 1);
  hipMemsetAsync(bufB, 0, (size_t)(N_NODES / 2) * HID * sizeof(float), stream);
  k_prop<7><<<cdiv(N_EDGES, 8), TB, 0, stream>>>(src, dst, w, bufA, bufB, N_EDGES, 1);
  k_cheb2<<<cdiv((size_t)(N_NODES / 2) * HID, TB), TB, 0, stream>>>(bufB, h2, (size_t)(N_NODES / 2) * HID);

  // out2 = relu(h2@W2[0] + Tb1@W2[1] + Tb2@W2[2] + b2)   [WMMA, 16x64 strips]
  {
    const unsigned waves = ((N_NODES / 2) / 16) * (HID / 64); // 6250
    k_cheb_gemm_relu<HID><<<cdiv(waves, 4), 128, 0, stream>>>(h2, bufA, bufB, W2, b2, out2, N_NODES / 2);
  }

  // ===== global mean pool + FC + log_softmax =====
  hipMemsetAsync(sums, 0, (NGRAPH * HID + NGRAPH) * sizeof(float), stream);
  k_pool<<<cdiv((size_t)(N_NODES / 2), 8), TB, 0, stream>>>(out2, batch, sums, cnts, N_NODES / 2);
  k_head<<<1, NGRAPH * NCLS, 0, stream>>>(sums, cnts, fcW, fcb, out);
}